// HeatEquation_43344809951328
// MI455X (gfx1250) — compile-verified
//
#include <hip/hip_runtime.h>
#include <stdint.h>

// ---------------------------------------------------------------------------
// Heat equation, 512x512, 255 recorded steps.
// Strategy (MI455X / gfx1250):
//   * 64 persistent workgroups x 256 threads (8 waves, wave32). WG g owns rows
//     [8g, 8g+8).
//   * Per step: TDM tensor_load_to_lds pulls the 10-row halo tile (20 KB) from
//     the previous frame into LDS; 5-point stencil computed in LDS; TDM
//     tensor_store_from_lds streams the 8-row result tile (16 KB) to d_out.
//   * d_out doubles as the state ping-pong: step s reads frame s-2 (or u0) and
//     writes frame s-1. Live state stays in the 192 MB L2.
//   * Diffusion map (4-Gaussian mixture of alpha) computed once into LDS.
//   * Device-wide sense-reversal barrier (agent-scope atomics) between steps.
// ---------------------------------------------------------------------------

#define NSIDE 512
#define NN    (NSIDE * NSIDE)
#define STEPS 255           // frames = states after steps 1..255
#define NWG   64
#define RPW   8             // rows per workgroup
#define TPB   256

typedef unsigned int u32x4 __attribute__((ext_vector_type(4)));
typedef int          i32x8 __attribute__((ext_vector_type(8)));
typedef int          i32x4 __attribute__((ext_vector_type(4)));

// ---- TDM descriptor builders (CDNA5 ISA ch.8, D# groups 0/1) --------------
__device__ __forceinline__ u32x4 tdm_g0(unsigned lds_byte_off, const void* gaddr) {
  unsigned long long ga = (unsigned long long)gaddr;
  u32x4 g0;
  g0.x = 1u;                                                   // count=1, user D#
  g0.y = lds_byte_off;                                         // lds_addr
  g0.z = (unsigned)ga;                                         // global_addr[31:0]
  g0.w = (unsigned)((ga >> 32) & 0x1FFFFFFull) | (2u << 30);   // addr[56:32], type=2
  return g0;
}

__device__ __forceinline__ i32x8 tdm_g1_2d(unsigned tensor_d0, unsigned tensor_d1,
                                           unsigned tile_d0, unsigned tile_d1,
                                           unsigned stride0_elems) {
  i32x8 g1;
  g1[0] = (int)(2u << 16);                                     // data_size = 4 B
  g1[1] = (int)(tensor_d0 << 16);                              // tensor_dim0[15:0]
  g1[2] = (int)((tensor_d0 >> 16) | (tensor_d1 << 16));        // dim0 hi | dim1 lo
  g1[3] = (int)((tensor_d1 >> 16) | (tile_d0 << 16));          // dim1 hi | tile_dim0
  g1[4] = (int)(tile_d1 & 0xFFFFu);                            // tile_dim1, tile_dim2=0
  g1[5] = (int)stride0_elems;                                  // tensor_dim0_stride lo
  g1[6] = 0;                                                   // stride0 hi | stride1 lo
  g1[7] = 0;
  return g1;
}

// ---- device-wide sense-reversal barrier -----------------------------------
__device__ __forceinline__ void grid_barrier(unsigned* bar, unsigned nwg) {
  __syncthreads();
  if (threadIdx.x == 0) {
    unsigned gen = __hip_atomic_load(&bar[1], __ATOMIC_ACQUIRE, __HIP_MEMORY_SCOPE_AGENT);
    unsigned arrived = __hip_atomic_fetch_add(&bar[0], 1u, __ATOMIC_ACQ_REL,
                                              __HIP_MEMORY_SCOPE_AGENT);
    if (arrived == nwg - 1u) {
      __hip_atomic_store(&bar[0], 0u, __ATOMIC_RELAXED, __HIP_MEMORY_SCOPE_AGENT);
      __hip_atomic_store(&bar[1], gen + 1u, __ATOMIC_RELEASE, __HIP_MEMORY_SCOPE_AGENT);
    } else {
      while (__hip_atomic_load(&bar[1], __ATOMIC_ACQUIRE,
                               __HIP_MEMORY_SCOPE_AGENT) == gen) {
        __builtin_amdgcn_s_sleep(2);
      }
    }
  }
  __syncthreads();
}

__global__ void heat_init_bar(unsigned* bar) { bar[0] = 0u; bar[1] = 0u; }

__global__ __launch_bounds__(TPB)
void heat_persistent(const float* __restrict__ u0, const float* __restrict__ alpha,
                     float* __restrict__ out, unsigned* __restrict__ bar) {
  __shared__ float lds_in[10 * NSIDE];    // halo tile, local row rr = global (8g-1)+rr
  __shared__ float lds_out[RPW * NSIDE];
  __shared__ float lds_dmap[RPW * NSIDE];

  const int g    = blockIdx.x;
  const int tid  = threadIdx.x;
  const int row0 = g * RPW;

  // -- diffusion map for our 8 rows, computed once -------------------------
  float amp[4], cx[4], cy[4], wd[4];
#pragma unroll
  for (int k = 0; k < 4; ++k) {
    amp[k] = alpha[k]; cx[k] = alpha[4 + k]; cy[k] = alpha[8 + k]; wd[k] = alpha[12 + k];
  }
  const float grid_step = 2.0f / 511.0f;   // linspace(-1, 1, 512)
  for (int idx = tid; idx < RPW * NSIDE; idx += TPB) {
    int r = row0 + (idx >> 9);
    int c = idx & (NSIDE - 1);
    float X = -1.0f + grid_step * (float)r;   // meshgrid 'ij': X varies with i
    float Y = -1.0f + grid_step * (float)c;
    float acc = 0.0f;
#pragma unroll
    for (int k = 0; k < 4; ++k) {
      float dxk = X - cx[k], dyk = Y - cy[k];
      acc += amp[k] * expf(-(dxk * dxk + dyk * dyk) / wd[k]);
    }
    lds_dmap[idx] = acc;
  }
  __syncthreads();

  const float dx2 = (float)((2.0 / 511.0) * (2.0 / 511.0));
  const float dt  = 1e-6f;
  const unsigned lds_in_off  = (unsigned)(unsigned long long)(uintptr_t)&lds_in[0];
  const unsigned lds_out_off = (unsigned)(unsigned long long)(uintptr_t)&lds_out[0];
  const i32x4 z4 = {0, 0, 0, 0};
  const i32x8 z8 = {0, 0, 0, 0, 0, 0, 0, 0};

  for (int s = 1; s <= STEPS; ++s) {
    const float* src = (s == 1) ? u0 : (out + (size_t)(s - 2) * NN);
    float*       dst = out + (size_t)(s - 1) * NN;

    // ---- TDM: halo tile (rows [8g-1, 8g+9) clamped) -> LDS --------------
    if (tid < 32) {   // wave 0 drives the DMA
      int lo = row0 - 1, skip = 0;
      if (lo < 0) { lo = 0; skip = 1; }            // top edge: land at local row 1
      unsigned nreq  = 10u - (unsigned)skip;       // bottom edge: OOB rows zero-fill
      unsigned avail = (unsigned)(NSIDE - lo);
      u32x4 g0 = tdm_g0(lds_in_off + (unsigned)skip * NSIDE * 4u,
                        src + (size_t)lo * NSIDE);
      i32x8 g1 = tdm_g1_2d(NSIDE, avail, NSIDE, nreq, NSIDE);
      __builtin_amdgcn_tensor_load_to_lds(g0, g1, z4, z4, z8, 16 /* scope: DEV */);
      __builtin_amdgcn_s_wait_tensorcnt(0);
    }
    __syncthreads();

    // ---- stencil in LDS --------------------------------------------------
    for (int idx = tid; idx < RPW * NSIDE; idx += TPB) {
      int rl = idx >> 9;
      int c  = idx & (NSIDE - 1);
      int r  = row0 + rl;
      float v = 0.0f;                              // Dirichlet boundary
      if (r > 0 && r < NSIDE - 1 && c > 0 && c < NSIDE - 1) {
        int rr = rl + 1;                           // local row in halo tile
        float tc  = lds_in[rr * NSIDE + c];
        float lap = (lds_in[(rr - 1) * NSIDE + c] + lds_in[(rr + 1) * NSIDE + c] +
                     lds_in[rr * NSIDE + c - 1]   + lds_in[rr * NSIDE + c + 1] -
                     4.0f * tc) / dx2;
        v = tc + dt * lds_dmap[idx] * lap;
      }
      lds_out[idx] = v;
    }
    __syncthreads();

    // ---- TDM: result tile LDS -> frame (device scope for x-WGP visibility)
    if (tid < 32) {
      u32x4 g0 = tdm_g0(lds_out_off, dst + (size_t)row0 * NSIDE);
      i32x8 g1 = tdm_g1_2d(NSIDE, RPW, NSIDE, RPW, NSIDE);
      __builtin_amdgcn_tensor_store_from_lds(g0, g1, z4, z4, z8, 16 /* scope: DEV */);
      __builtin_amdgcn_s_wait_tensorcnt(0);
    }

    grid_barrier(bar, (unsigned)gridDim.x);
  }
}

extern "C" void kernel_launch(void* const* d_in, const int* in_sizes, int n_in,
                              void* d_out, int out_size, void* d_ws, size_t ws_size,
                              hipStream_t stream) {
  const float* u0    = (const float*)d_in[0];   // [512*512] f32
  const float* alpha = (const float*)d_in[1];   // [16] f32
  float*       out   = (float*)d_out;           // [255*512*512] f32
  unsigned*    bar   = (unsigned*)d_ws;         // 8 bytes of scratch

  heat_init_bar<<<1, 1, 0, stream>>>(bar);
  heat_persistent<<<NWG, TPB, 0, stream>>>(u0, alpha, out, bar);
}